// SSAM_17721035063750
// MI455X (gfx1250) — compile-verified
//
#include <hip/hip_runtime.h>
#include <stdint.h>

// Problem constants (match reference)
#define BB      32
#define CC      512
#define HWTOT   4096
#define KCLS    17               // 16 classes via WMMA + class 16 via VALU
#define CHUNK   64               // hw columns per sub-tile
#define SUBT    4                // sub-tiles per block (256 hw columns per block)
#define PITCH   68               // LDS row pitch in dwords (64 data + 4 pad)
#define NBLOCK  (BB * (HWTOT / (CHUNK * SUBT)))   // 32 * 16 = 512 blocks

#define XS_FLOATS (CC * PITCH)         // one X buffer: 34816 floats (139264 B)
#define SS_FLOATS (KCLS * PITCH)       // S tile: 17 rows

typedef float v2f __attribute__((ext_vector_type(2)));
typedef float v8f __attribute__((ext_vector_type(8)));
typedef unsigned int u32x4 __attribute__((ext_vector_type(4)));
typedef unsigned int u32x8 __attribute__((ext_vector_type(8)));

__device__ __forceinline__ float fast_sigmoid(float v) {
    // v_exp_f32 + v_rcp_f32, no IEEE-division expansion
    return __builtin_amdgcn_rcpf(1.0f + __expf(-v));
}

// ---------------------------------------------------------------------------
// TDM: async load of a 512x64 fp32 tile (row stride HWTOT) into LDS, pitch 68
// ---------------------------------------------------------------------------
__device__ __forceinline__ void tdm_load_tile(const float* gbase, unsigned lds_off) {
    const unsigned long long gaddr = (unsigned long long)(uintptr_t)gbase;
    u32x4 g0;
    g0[0] = 1u;                                      // count=1, user descriptor
    g0[1] = lds_off;                                 // lds_addr (bytes)
    g0[2] = (unsigned)gaddr;                         // global_addr[31:0]
    g0[3] = ((unsigned)(gaddr >> 32) & 0x01FFFFFFu)  // global_addr[56:32]
          | 0x80000000u;                             // type=2 ("image")
    u32x8 g1;
    g1[0] = (2u << 16)        // data_size = 4 bytes
          | (1u << 20)        // pad_enable
          | (5u << 22)        // pad_interval: 64 DWORDs
          | (3u << 25);       // pad_amount:   4 DWORDs  -> pitch 68
    g1[1] = ((unsigned)HWTOT & 0xFFFFu) << 16;              // tensor_dim0 lo16
    g1[2] = ((unsigned)HWTOT >> 16) | ((unsigned)CC << 16); // dim0 hi16 | tensor_dim1 lo16
    g1[3] = ((unsigned)CC >> 16) | ((unsigned)CHUNK << 16); // dim1 hi16 | tile_dim0
    g1[4] = (unsigned)CC;                                   // tile_dim1 | tile_dim2=0
    g1[5] = (unsigned)HWTOT;                                // tensor_dim0_stride lo32
    g1[6] = 0u;
    g1[7] = 0u;
    asm volatile("tensor_load_to_lds %0, %1" :: "s"(g0), "s"(g1) : "memory");
}

// ---------------------------------------------------------------------------
// Main fused kernel: one block per (batch, 256-hw slab), 4 sub-tiles,
// double-buffered TDM loads overlapping fp32 WMMA compute.
// ---------------------------------------------------------------------------
__global__ void __launch_bounds__(128)
ssam_main_kernel(const float* __restrict__ x,
                 const float* __restrict__ w,
                 const float* __restrict__ bias,
                 float* __restrict__ xsum,
                 float* __restrict__ sf) {
    extern __shared__ float smem[];
    float* Ss = smem + 2 * XS_FLOATS;         // 17 x PITCH

    const int bidx   = blockIdx.x >> 4;       // batch image (32)
    const int slab   = blockIdx.x & 15;       // 16 slabs of 256 hw columns
    const int hwbase = slab * (CHUNK * SUBT);

    const int tid  = threadIdx.x;
    const int lane = tid & 31;
    const int wave = tid >> 5;                // 0..3
    const int lm   = lane & 15;
    const int half = lane >> 4;               // 0: lanes 0-15, 1: lanes 16-31

    const unsigned lds_base = (unsigned)(uintptr_t)(void*)smem;
    const float* xslab = x + (size_t)bidx * CC * HWTOT + hwbase;
    const float b16 = bias[16];
    const float* w16 = w + 16 * CC;

    // Persistent accumulators across the 4 sub-tiles
    v8f   sfacc[8];                           // sf rows 0..15, this wave's 128 channels
    float sf16a[4];                           // sf row 16, channels tid + 128*j
    float xs4[4];                             // xsum partials, channels tid + 128*j
    #pragma unroll
    for (int t = 0; t < 8; ++t) sfacc[t] = (v8f){};
    #pragma unroll
    for (int j = 0; j < 4; ++j) { sf16a[j] = 0.0f; xs4[j] = 0.0f; }

    // Prime the pipeline: load sub-tile 0 into buffer 0
    if (wave == 0) tdm_load_tile(xslab, lds_base);

    for (int s = 0; s < SUBT; ++s) {
        const float* Xs = smem + (unsigned)(s & 1) * XS_FLOATS;   // current buffer

        if (wave == 0) __builtin_amdgcn_s_wait_tensorcnt(0);  // tile s resident
        __syncthreads();

        // Kick off the next sub-tile's DMA into the other buffer (overlaps compute)
        if (wave == 0 && s + 1 < SUBT)
            tdm_load_tile(xslab + (s + 1) * CHUNK,
                          lds_base + (unsigned)((s + 1) & 1) * (XS_FLOATS * 4u));

        // ---- xsum partials (for g): each thread owns 4 channels ----
        #pragma unroll
        for (int j = 0; j < 4; ++j) {
            const float* row = Xs + (tid + 128 * j) * PITCH;
            float a = 0.0f;
            #pragma unroll 8
            for (int i = 0; i < CHUNK; ++i) a += row[i];
            xs4[j] += a;
        }

        // ---- class-16 cam row via VALU (threads 0..63, one hw column each) ----
        if (tid < CHUNK) {
            float c16 = b16;
            #pragma unroll 8
            for (int c = 0; c < CC; ++c) c16 += w16[c] * Xs[c * PITCH + tid];
            Ss[16 * PITCH + tid] = fast_sigmoid(c16);
        }

        // ---- GEMM1: cam tile (classes 0..15 x 16 hw per wave), K over C ----
        const int ncol = wave * 16;
        v8f acc0;
        #pragma unroll
        for (int v = 0; v < 8; ++v) acc0[v] = bias[v + half * 8];

        #pragma unroll 8
        for (int c0 = 0; c0 < CC; c0 += 4) {
            const int kk = c0 + (half << 1);
            float2 a0 = *(const float2*)(w + (size_t)lm * CC + kk);      // A: W rows 0..15
            float bx = Xs[(kk)     * PITCH + ncol + lm];                 // B: X rows kk,kk+1
            float by = Xs[(kk + 1) * PITCH + ncol + lm];
            v2f A0 = {a0.x, a0.y};
            v2f Bv = {bx, by};
            acc0 = __builtin_amdgcn_wmma_f32_16x16x4_f32(false, A0, false, Bv,
                                                         (short)0, acc0, false, false);
        }

        // ---- sigmoid -> S tile (rows 0..15) ----
        #pragma unroll
        for (int v = 0; v < 8; ++v) {
            int cls = v + half * 8;
            Ss[cls * PITCH + ncol + lm] = fast_sigmoid(acc0[v]);
        }
        __syncthreads();

        // ---- GEMM2: sf += S(16 x 64) @ Xtile^T(64 x 128-per-wave) ----
        #pragma unroll
        for (int t = 0; t < 8; ++t) {
            const int cbase = (wave * 8 + t) * 16;
            v8f acc = sfacc[t];
            #pragma unroll
            for (int k0 = 0; k0 < CHUNK; k0 += 4) {
                const int kk = k0 + (half << 1);
                float2 a = *(const float2*)(Ss + lm * PITCH + kk);            // A: S rows
                float2 b = *(const float2*)(Xs + (cbase + lm) * PITCH + kk);  // B: X^T
                v2f A = {a.x, a.y};
                v2f Bv = {b.x, b.y};
                acc = __builtin_amdgcn_wmma_f32_16x16x4_f32(false, A, false, Bv,
                                                            (short)0, acc, false, false);
            }
            sfacc[t] = acc;
        }

        // ---- sf row 16 via VALU: 4 channels per thread ----
        const float* s16row = Ss + 16 * PITCH;
        #pragma unroll
        for (int j = 0; j < 4; ++j) {
            const float* row = Xs + (tid + 128 * j) * PITCH;
            float a = 0.0f;
            #pragma unroll 8
            for (int i = 0; i < CHUNK; ++i) a += s16row[i] * row[i];
            sf16a[j] += a;
        }
        __syncthreads();   // everyone done with Xs / Ss before next iteration
    }

    // ---- flush accumulators (one atomic pass per block) ----
    const float inv = 1.0f / (float)HWTOT;
    #pragma unroll
    for (int t = 0; t < 8; ++t) {
        const int cbase = (wave * 8 + t) * 16;
        #pragma unroll
        for (int v = 0; v < 8; ++v) {
            int cls = v + half * 8;
            atomicAdd(&sf[((size_t)bidx * KCLS + cls) * CC + cbase + lm],
                      sfacc[t][v] * inv);
        }
    }
    #pragma unroll
    for (int j = 0; j < 4; ++j) {
        int c = tid + 128 * j;
        atomicAdd(&sf[((size_t)bidx * KCLS + 16) * CC + c], sf16a[j] * inv);
        atomicAdd(&xsum[bidx * CC + c], xs4[j]);
    }
}

// ---------------------------------------------------------------------------
// g = (xsum / HW) @ W^T + bias     (B*K = 544 dot products of length 512)
// ---------------------------------------------------------------------------
__global__ void ssam_g_kernel(const float* __restrict__ xsum,
                              const float* __restrict__ w,
                              const float* __restrict__ bias,
                              float* __restrict__ g) {
    int t = blockIdx.x * blockDim.x + threadIdx.x;
    if (t >= BB * KCLS) return;
    int b = t / KCLS, k = t % KCLS;
    const float* xs = xsum + b * CC;
    const float* wr = w + k * CC;
    float s = 0.0f;
    #pragma unroll 8
    for (int c = 0; c < CC; ++c) s += xs[c] * wr[c];
    g[t] = s * (1.0f / (float)HWTOT) + bias[k];
}

// ---------------------------------------------------------------------------
extern "C" void kernel_launch(void* const* d_in, const int* in_sizes, int n_in,
                              void* d_out, int out_size, void* d_ws, size_t ws_size,
                              hipStream_t stream) {
    const float* x    = (const float*)d_in[0];   // (32, 512, 64, 64)
    const float* w    = (const float*)d_in[1];   // (17, 512)
    const float* bias = (const float*)d_in[2];   // (17,)

    float* out = (float*)d_out;
    float* g_out  = out;                 // 32*17
    float* sf_out = out + BB * KCLS;     // 32*17*512

    float* xsum = (float*)d_ws;          // BB*CC floats

    // zero accumulation targets (graph-capture safe)
    (void)hipMemsetAsync(d_out, 0, (size_t)out_size * sizeof(float), stream);
    (void)hipMemsetAsync(xsum, 0, (size_t)BB * CC * sizeof(float), stream);

    // fused cam/sigmoid/sf kernel: 512 blocks, double-buffered TDM + WMMA
    const size_t smem_bytes = (size_t)(2 * XS_FLOATS + SS_FLOATS) * sizeof(float);
    ssam_main_kernel<<<dim3(NBLOCK), dim3(128), smem_bytes, stream>>>(
        x, w, bias, xsum, sf_out);

    // g from xsum
    ssam_g_kernel<<<(BB * KCLS + 127) / 128, 128, 0, stream>>>(xsum, w, bias, g_out);
}